// EdgeBlock_29119878266986
// MI455X (gfx1250) — compile-verified
//
#include <hip/hip_runtime.h>

typedef __attribute__((ext_vector_type(2))) float v2f;
typedef __attribute__((ext_vector_type(8))) float v8f;

#define DMODEL   256
#define KTOT     768          // 3 * DMODEL
#define NSTRIPE  64           // output columns handled per block (blockIdx.y selects stripe)
#define NT       4            // 16-wide n-tiles per stripe
#define NTHREADS 512          // 16 wave32 waves -> 4 waves per SIMD32 (latency hiding)
#define NWAVES   (NTHREADS / 32)
#define MTILE    (NWAVES * 16)   // 256 edges per block per M iteration

// Out[e, :] = concat(edges[e], nodes[recv[e]], nodes[send[e]]) @ W   (f32, exact)
// One wave computes a 16-row x 64-col output strip as 4 WMMA 16x16 tiles,
// accumulating over K=768 in steps of 4 via V_WMMA_F32_16X16X4_F32.
__global__ __launch_bounds__(NTHREADS)
void edgeblock_wmma_f32(const float* __restrict__ nodes,
                        const float* __restrict__ edges,
                        const int*   __restrict__ receivers,
                        const int*   __restrict__ senders,
                        const float* __restrict__ W,
                        float*       __restrict__ out,
                        int mTiles)
{
    // W stripe in LDS, pair-interleaved so each lane's B fragment is a contiguous b64:
    // element (k, c) lives at float offset ((k>>1)*NSTRIPE + c)*2 + (k&1)
    __shared__ float wlds[KTOT * NSTRIPE];   // 768*64*4 = 192 KB (<= 320 KB WGP LDS)

    const int tid  = threadIdx.x;
    const int wave = tid >> 5;
    const int lane = tid & 31;
    const int n    = lane & 15;    // row (A) / column (B,C,D) index within tile
    const int half = lane >> 4;    // K-pair selector for A/B fragments, M-half for C/D
    const int nOff = blockIdx.y * NSTRIPE;

    // ---- cooperative load of the W column-stripe into LDS (once per block) ----
    for (int idx = tid; idx < KTOT * NSTRIPE; idx += NTHREADS) {
        const int k = idx >> 6;          // NSTRIPE == 64
        const int c = idx & (NSTRIPE - 1);
        wlds[((((k >> 1) * NSTRIPE) + c) << 1) | (k & 1)] = W[k * DMODEL + nOff + c];
    }
    __syncthreads();   // LDS is read-only below; no barriers needed in the hot loop

    const v8f zero = {};

    for (int mt = blockIdx.x; mt < mTiles; mt += gridDim.x) {
        // This lane supplies A-matrix row m = n of this wave's 16-edge tile.
        const int e = mt * MTILE + wave * 16 + n;

        const float* baseA[3];
        baseA[0] = edges + (size_t)e * DMODEL;                  // K segment [0,256)
        baseA[1] = nodes + (size_t)receivers[e] * DMODEL;       // K segment [256,512)
        baseA[2] = nodes + (size_t)senders[e]   * DMODEL;       // K segment [512,768)

        v8f acc[NT];
        #pragma unroll
        for (int t = 0; t < NT; ++t) acc[t] = zero;

        #pragma unroll
        for (int s = 0; s < 3; ++s) {
            const float* pa = baseA[s] + 2 * half;   // lanes 16-31 hold K = k0+2, k0+3
            const int segBase = s * DMODEL;
            #pragma unroll 4
            for (int k0 = 0; k0 < DMODEL; k0 += 4) {
                const v2f a = *(const v2f*)(pa + k0);          // A[m][k0+2h], A[m][k0+2h+1]
                const int p = ((segBase + k0) >> 1) + half;    // W row-pair index
                #pragma unroll
                for (int t = 0; t < NT; ++t) {
                    const v2f b = *(const v2f*)&wlds[((p * NSTRIPE) + (t * 16 + n)) << 1];
                    acc[t] = __builtin_amdgcn_wmma_f32_16x16x4_f32(
                        /*neg_a=*/false, a, /*neg_b=*/false, b,
                        /*c_mod=*/(short)0, acc[t],
                        /*reuse_a=*/false, /*reuse_b=*/false);
                }
            }
        }

        // ---- store: acc[t] VGPR r holds element (m = r + 8*half, col = nOff + t*16 + n)
        #pragma unroll
        for (int t = 0; t < NT; ++t) {
            const int col = nOff + t * 16 + n;
            #pragma unroll
            for (int r = 0; r < 8; ++r) {
                const int m  = r + 8 * half;
                const int er = mt * MTILE + wave * 16 + m;
                out[(size_t)er * DMODEL + col] = acc[t][r];
            }
        }
    }
}

extern "C" void kernel_launch(void* const* d_in, const int* in_sizes, int n_in,
                              void* d_out, int out_size, void* d_ws, size_t ws_size,
                              hipStream_t stream) {
    const float* nodes     = (const float*)d_in[0];
    const float* edges     = (const float*)d_in[1];
    const int*   receivers = (const int*)  d_in[2];
    const int*   senders   = (const int*)  d_in[3];
    const float* W         = (const float*)d_in[4];
    float*       out       = (float*)d_out;

    const int nEdges = in_sizes[1] / DMODEL;      // 160000
    const int mTiles = nEdges / MTILE;            // 625

    int px = mTiles < 160 ? mTiles : 160;         // grid-stride over M amortizes W->LDS load
    dim3 grid(px, DMODEL / NSTRIPE, 1);           // (160, 4)
    dim3 block(NTHREADS, 1, 1);                   // 16 wave32 waves
    edgeblock_wmma_f32<<<grid, block, 0, stream>>>(nodes, edges, receivers, senders,
                                                   W, out, mTiles);
}